// CSPN_25074019074065
// MI455X (gfx1250) — compile-verified
//
#include <hip/hip_runtime.h>

// ---------------------------------------------------------------------------
// CSPN propagation step, MI455X (gfx1250).
// Bandwidth-bound stencil: ~185 MB minimum traffic -> ~8 us at 23.3 TB/s.
// Strategy: async global->LDS DMA staging of the haloed affinity tile
// (ASYNCcnt path, CDNA5-specific), in-place normalization in LDS (reuses the
// per-pixel 1/abs_sum across its 8 consumers), then an 8-neighbor LDS gather.
// ---------------------------------------------------------------------------

#ifndef __has_builtin
#define __has_builtin(x) 0
#endif

#if __has_builtin(__builtin_amdgcn_global_load_async_to_lds_b32)
#define HAVE_ASYNC_LDS 1
#else
#define HAVE_ASYNC_LDS 0
#endif

// Clang prints addrspace(1) as `__device__` and addrspace(3) as `__shared__`
// in HIP diagnostics; the async builtin takes int* in those address spaces.
typedef __attribute__((address_space(1))) int gint_t;   // global
typedef __attribute__((address_space(3))) int lint_t;   // LDS

namespace {
constexpr int IMG     = 256;                 // H == W == 256
constexpr int NCH     = 8;                   // affinity channels
constexpr int TX      = 32;                  // tile width  (one wave per row)
constexpr int TY      = 8;                   // tile height (8 waves / block)
constexpr int HALO_W  = TX + 2;              // 34
constexpr int HALO_H  = TY + 2;              // 10
constexpr int N_HALO  = HALO_W * HALO_H;     // 340
constexpr int NTHREADS = TX * TY;            // 256 threads = 8 wave32s
}

__global__ __launch_bounds__(NTHREADS) void cspn_kernel(
    const float* __restrict__ aff,   // [B, 8, 256, 256]
    const float* __restrict__ cur,   // [B, 1, 256, 256]
    const float* __restrict__ coa,   // [B, 1, 256, 256]
    float* __restrict__ out)         // [B, 1, 256, 256]
{
  // 9 planes of 340 floats = 12.2 KB LDS (320 KB/WGP -> high occupancy).
  __shared__ float s_a[NCH][N_HALO];   // normalized affinity (in place)
  __shared__ float s_rs[N_HALO];       // 1 - raw_sum (center term)

  const int tid = threadIdx.x;
  const int tx  = tid & 31;
  const int ty  = tid >> 5;
  const int gx0 = blockIdx.x * TX;
  const int gy0 = blockIdx.y * TY;
  const int b   = blockIdx.z;

  const size_t plane = (size_t)IMG * IMG;
  const float* affb  = aff + (size_t)b * NCH * plane;

  // ---- Phase 1: stage haloed raw-affinity tile into LDS (async DMA). ------
  for (int p = tid; p < N_HALO; p += NTHREADS) {
    const int lyy = p / HALO_W;
    const int lxx = p - lyy * HALO_W;
    int gy = gy0 - 1 + lyy;
    int gx = gx0 - 1 + lxx;
    // Clamp; out-of-image halo entries are masked at consumption time.
    gy = gy < 0 ? 0 : (gy > IMG - 1 ? IMG - 1 : gy);
    gx = gx < 0 ? 0 : (gx > IMG - 1 ? IMG - 1 : gx);
    const float* gp = affb + (size_t)gy * IMG + gx;
#pragma unroll
    for (int c = 0; c < NCH; ++c) {
#if HAVE_ASYNC_LDS
      __builtin_amdgcn_global_load_async_to_lds_b32(
          (gint_t*)(gp + (size_t)c * plane),
          (lint_t*)&s_a[c][p],
          /*imm offset=*/0, /*cpol=*/0);
#else
      s_a[c][p] = gp[(size_t)c * plane];
#endif
    }
  }
#if HAVE_ASYNC_LDS
#if __has_builtin(__builtin_amdgcn_s_wait_asynccnt)
  __builtin_amdgcn_s_wait_asynccnt(0);
#else
  asm volatile("s_wait_asynccnt 0" ::: "memory");
#endif
#endif
  __syncthreads();

  // ---- Phase 2: per-pixel normalization, in place in LDS. -----------------
  for (int p = tid; p < N_HALO; p += NTHREADS) {
    float v[NCH];
    float asum = 0.0f, rsum = 0.0f;
#pragma unroll
    for (int c = 0; c < NCH; ++c) {
      v[c] = s_a[c][p];
      asum += fabsf(v[c]);
      rsum += v[c];
    }
    const float r = 1.0f / asum;
#pragma unroll
    for (int c = 0; c < NCH; ++c) s_a[c][p] = v[c] * r;
    s_rs[p] = 1.0f - rsum;
  }
  __syncthreads();

  // ---- Phase 3: 8-neighbor gather + center term. --------------------------
  const int x  = gx0 + tx;
  const int y  = gy0 + ty;
  const int lx = tx + 1;
  const int ly = ty + 1;
  const size_t pix = (size_t)b * plane + (size_t)y * IMG + x;

  const float c_cur = cur[pix];
  const float c_coa = coa[pix];

  // out[y,x] = cur[y,x] * sum_{k!=4} a_k[y-dy_k, x-dx_k]  +  (1-raw_sum)[y,x]*coa[y,x]
  float acc = 0.0f;
#pragma unroll
  for (int k = 0; k < 9; ++k) {
    if (k == 4) continue;                       // center handled below
    const int dy  = k / 3 - 1;
    const int dx  = k % 3 - 1;
    const int chn = (k < 4) ? k : (k - 1);      // new_aff ch -> affinity ch
    const int ny  = y - dy;
    const int nx  = x - dx;
    const bool valid = ((unsigned)ny < (unsigned)IMG) &&
                       ((unsigned)nx < (unsigned)IMG);
    const float wv = s_a[chn][(ly - dy) * HALO_W + (lx - dx)];
    acc += valid ? wv : 0.0f;
  }
  out[pix] = c_cur * acc + s_rs[ly * HALO_W + lx] * c_coa;
}

extern "C" void kernel_launch(void* const* d_in, const int* in_sizes, int n_in,
                              void* d_out, int out_size, void* d_ws, size_t ws_size,
                              hipStream_t stream) {
  (void)n_in; (void)out_size; (void)d_ws; (void)ws_size;
  const float* aff = (const float*)d_in[0];
  const float* cur = (const float*)d_in[1];
  const float* coa = (const float*)d_in[2];
  float* out = (float*)d_out;

  const int B = in_sizes[1] / (IMG * IMG);     // cur is [B,1,256,256]
  dim3 grid(IMG / TX, IMG / TY, B);
  cspn_kernel<<<grid, dim3(NTHREADS), 0, stream>>>(aff, cur, coa, out);
}